// ModelHMMTorch_28561532518941
// MI455X (gfx1250) — compile-verified
//
#include <hip/hip_runtime.h>

// CDNA5 (gfx1250) Kalman/RTS smoother.
// fp32 WMMA (V_WMMA_F32_16X16X4_F32) + Tensor Data Mover staging of GEMM
// operands into LDS (double-buffered); the MMA inner loop also reads some operands from global.

typedef float v2f __attribute__((ext_vector_type(2)));
typedef float v8f __attribute__((ext_vector_type(8)));
typedef unsigned int u32x4 __attribute__((ext_vector_type(4)));
typedef int i32x8 __attribute__((ext_vector_type(8)));
typedef int i32x4 __attribute__((ext_vector_type(4)));

constexpr int kT = 512, kB = 256, kN = 128, kD = 64;
constexpr int kNN = kN * kN;   // 16384
constexpr int kND = kN * kD;   // 8192
constexpr int kDD = kD * kD;   // 4096
constexpr int kBT = 32;        // batch rows per block in mean kernels

// ---------------------------------------------------------------------------
// TDM: issue a 2D tile load (rows x cols fp32, row stride ld elements),
// global -> LDS. Must be reached by exactly ONE wave (TDM ignores EXEC but
// issues once per wave). Completion: tdm_wait() in the issuing wave, then a
// workgroup barrier publishes the LDS data to all waves.
// D# layout per CDNA5 ISA ch.8 (group0 128b, group1 256b; groups 2/3 unused
// for 2D tensors). This toolchain exposes the 6-arg builtin
// (u32x4 g0, i32x8 g1, i32x4 g2, i32x4 g3, i32x8 extra, i32 cpol).
// ---------------------------------------------------------------------------
__device__ __forceinline__ void tdm_load_2d(const float* gsrc, void* ldst,
                                            int rows, int cols, int ld)
{
  const unsigned long long ga = (unsigned long long)(size_t)gsrc;
  const unsigned lds = (unsigned)(size_t)ldst;  // flat LDS aperture: low 32 bits
  u32x4 g0;
  g0[0] = 1u;                                   // count=1 (valid user D#)
  g0[1] = lds;                                  // lds_addr (bytes)
  g0[2] = (unsigned)ga;                         // global_addr[31:0]
  g0[3] = (unsigned)((ga >> 32) & 0x01FFFFFFu) | (2u << 30);  // [56:32] | type=2
  i32x8 g1;
  g1[0] = (2 << 16);                                           // data_size=4B
  g1[1] = (cols & 0xFFFF) << 16;                               // tensor_dim0 lo16
  g1[2] = ((rows & 0xFFFF) << 16) | ((cols >> 16) & 0xFFFF);   // dim1 lo | dim0 hi
  g1[3] = ((cols & 0xFFFF) << 16) | ((rows >> 16) & 0xFFFF);   // tile_dim0 | dim1 hi
  g1[4] = (rows & 0xFFFF);                                     // tile_dim1, tile_dim2=0
  g1[5] = ld;                                                  // tensor_dim0_stride
  g1[6] = 0;
  g1[7] = 0;
  i32x4 z4 = {0, 0, 0, 0};
  i32x8 z8 = {0, 0, 0, 0, 0, 0, 0, 0};
  __builtin_amdgcn_tensor_load_to_lds(g0, g1, z4, z4, z8, 0);
}

__device__ __forceinline__ void tdm_wait() {
  __builtin_amdgcn_s_wait_tensorcnt(0);         // s_wait_tensorcnt 0
}

// ---------------------------------------------------------------------------
// D = alpha * (A @ B[^T]) + C     (C may be nullptr)
// Row-major; M multiple of 16, Nn multiple of 32, K multiple of 4.
// Each wave owns a PAIR of adjacent 16x16 output tiles: one A fragment feeds
// two independent WMMA chains (latency hiding, half the A-fragment loads).
// Fragment layouts per CDNA5 ISA 7.12.2.
// ---------------------------------------------------------------------------
__device__ __forceinline__ void wmma_gemm_f32(
    const float* __restrict__ A, int lda,
    const float* __restrict__ B, int ldb, bool transB,
    const float* __restrict__ C, int ldc, float alpha,
    float* __restrict__ Dm, int ldd,
    int M, int Nn, int K, int lane, int wave, int nwaves)
{
  const int tilesN2 = Nn >> 5;                  // pairs of 16-wide tiles
  const int npairs = (M >> 4) * tilesN2;
  const int col = lane & 15;
  const int hi  = lane >> 4;
  for (int p = wave; p < npairs; p += nwaves) {
    const int tm = (p / tilesN2) << 4;
    const int tn = (p % tilesN2) << 5;
    v8f acc0 = {0.f, 0.f, 0.f, 0.f, 0.f, 0.f, 0.f, 0.f};
    v8f acc1 = {0.f, 0.f, 0.f, 0.f, 0.f, 0.f, 0.f, 0.f};
    const float* aRow = A + (tm + col) * lda;
    const float* bc0 = transB ? (B + (tn + col) * ldb) : nullptr;
    const float* bc1 = transB ? (B + (tn + 16 + col) * ldb) : nullptr;
    for (int k0 = 0; k0 < K; k0 += 4) {
      const int ka = k0 + (hi << 1);
      v2f af, b0, b1;
      af.x = aRow[ka];
      af.y = aRow[ka + 1];
      if (!transB) {
        b0.x = B[ka * ldb + tn + col];
        b0.y = B[(ka + 1) * ldb + tn + col];
        b1.x = B[ka * ldb + tn + 16 + col];
        b1.y = B[(ka + 1) * ldb + tn + 16 + col];
      } else {
        b0.x = bc0[ka];
        b0.y = bc0[ka + 1];
        b1.x = bc1[ka];
        b1.y = bc1[ka + 1];
      }
      acc0 = __builtin_amdgcn_wmma_f32_16x16x4_f32(false, af, false, b0,
                                                   (short)0, acc0, false, false);
      acc1 = __builtin_amdgcn_wmma_f32_16x16x4_f32(false, af, false, b1,
                                                   (short)0, acc1, false, false);
    }
#pragma unroll
    for (int v = 0; v < 8; ++v) {
      const int r = tm + v + (hi << 3);
      float o0 = alpha * acc0[v];
      float o1 = alpha * acc1[v];
      if (C) {
        o0 += C[r * ldc + tn + col];
        o1 += C[r * ldc + tn + 16 + col];
      }
      Dm[r * ldd + tn + col] = o0;
      Dm[r * ldd + tn + 16 + col] = o1;
    }
  }
}

// ---------------------------------------------------------------------------
// SPD inverse via Newton-Schulz: X_{k+1} = X_k (2I - S X_k), X_0 = I/trace(S).
// Pure-GEMM replacement for pinvh (hermitian PSD). `iters` must be even.
// ---------------------------------------------------------------------------
__device__ void spd_inverse(const float* __restrict__ S, float* X, float* T1,
                            float* X2, float* red, int n, int iters,
                            int tid, int nth, int lane, int wave, int nwaves)
{
  if (tid == 0) *red = 0.f;
  __syncthreads();
  if (tid < n) atomicAdd(red, S[tid * n + tid]);   // ds_add_f32
  __syncthreads();
  const float c = 1.0f / (*red);
  for (int i = tid; i < n * n; i += nth)
    X[i] = ((i / n) == (i % n)) ? c : 0.f;
  __syncthreads();
  float* Xa = X;
  float* Xb = X2;
  for (int it = 0; it < iters; ++it) {
    wmma_gemm_f32(S, n, Xa, n, false, nullptr, 0, 1.f, T1, n, n, n, n,
                  lane, wave, nwaves);
    __syncthreads();
    for (int i = tid; i < n * n; i += nth) {
      float v = -T1[i];
      if ((i / n) == (i % n)) v += 2.f;
      T1[i] = v;
    }
    __syncthreads();
    wmma_gemm_f32(Xa, n, T1, n, false, nullptr, 0, 1.f, Xb, n, n, n, n,
                  lane, wave, nwaves);
    __syncthreads();
    float* t = Xa; Xa = Xb; Xb = t;
  }
  // iters even -> result already in X
}

// ---------------------------------------------------------------------------
// Kernel 1: forward covariance filter (sequential in T, one workgroup).
// LDS: bufPred | bufT (tmp / PBt+G) | bufP (ptt / sig+X+T1+X2) | AS = 256 KB.
// A_t staged via TDM; the DMA for the next step overlaps PBt/sig/NS work.
// ---------------------------------------------------------------------------
__global__ void cov_forward_kernel(
    const float* __restrict__ a_g,   // (T-1,N,N)
    const float* __restrict__ q_g,   // (T-1,N,N)
    const float* __restrict__ b_g,   // (D,N)
    const float* __restrict__ r_g,   // (D,D)
    const float* __restrict__ p0_g,  // (N,N)
    float* __restrict__ ws_pttm1,    // (T,N,N) predicted covs
    float* __restrict__ ws_ptt,      // (T,N,N) filtered covs
    float* __restrict__ ws_g,        // (T,N,D) Kalman gains
    float* __restrict__ out_sig)     // (T,D,D)
{
  extern __shared__ float smem[];
  float* bufPred = smem;            // N*N : p_pred
  float* bufT    = smem + kNN;      // N*N : tmp / [PBt | G]
  float* bufP    = smem + 2 * kNN;  // N*N : ptt_prev / [sig X T1 X2]
  float* AS      = smem + 3 * kNN;  // N*N : TDM-staged A_t
  float* red     = smem + 4 * kNN;
  float* PBt = bufT;
  float* G   = bufT + kND;
  float* sig = bufP;
  float* X   = bufP + kDD;
  float* T1  = bufP + 2 * kDD;
  float* X2  = bufP + 3 * kDD;

  const int tid = threadIdx.x, nth = blockDim.x;
  const int lane = tid & 31, wave = tid >> 5, nwaves = nth >> 5;

  for (int t = 0; t < kT; ++t) {
    if (t == 0) {
      for (int i = tid; i < kNN; i += nth) bufPred[i] = p0_g[i];
      __syncthreads();
      if (wave == 0) tdm_load_2d(a_g, AS, kN, kN, kN);   // A_0 for t=1
    } else {
      if (wave == 0) tdm_wait();                          // AS = A_{t-1} ready
      __syncthreads();
      // tmp = A @ ptt_prev
      wmma_gemm_f32(AS, kN, bufP, kN, false, nullptr, 0, 1.f, bufT, kN,
                    kN, kN, kN, lane, wave, nwaves);
      __syncthreads();
      // p_pred = tmp @ A^T + Q
      wmma_gemm_f32(bufT, kN, AS, kN, true, q_g + (size_t)(t - 1) * kNN, kN,
                    1.f, bufPred, kN, kN, kN, kN, lane, wave, nwaves);
      __syncthreads();
      if (t < kT - 1 && wave == 0)                        // prefetch A_t
        tdm_load_2d(a_g + (size_t)t * kNN, AS, kN, kN, kN);
    }
    __syncthreads();
    for (int i = tid; i < kNN; i += nth)
      ws_pttm1[(size_t)t * kNN + i] = bufPred[i];
    __syncthreads();
    // PBt = p_pred @ B^T   (N x D)
    wmma_gemm_f32(bufPred, kN, b_g, kN, true, nullptr, 0, 1.f, PBt, kD,
                  kN, kD, kN, lane, wave, nwaves);
    __syncthreads();
    // sig = B @ PBt + R    (D x D)
    wmma_gemm_f32(b_g, kN, PBt, kD, false, r_g, kD, 1.f, sig, kD,
                  kD, kD, kN, lane, wave, nwaves);
    __syncthreads();
    for (int i = tid; i < kDD; i += nth)
      out_sig[(size_t)t * kDD + i] = sig[i];
    __syncthreads();
    spd_inverse(sig, X, T1, X2, red, kD, 22, tid, nth, lane, wave, nwaves);
    // G = PBt @ sig^-1     (N x D)
    wmma_gemm_f32(PBt, kD, X, kD, false, nullptr, 0, 1.f, G, kD,
                  kN, kD, kD, lane, wave, nwaves);
    __syncthreads();
    for (int i = tid; i < kND; i += nth)
      ws_g[(size_t)t * kND + i] = G[i];
    __syncthreads();
    // ptt = p_pred - G @ PBt^T   (symmetry: B@p_pred = PBt^T)
    wmma_gemm_f32(G, kD, PBt, kD, true, bufPred, kN, -1.f, bufP, kN,
                  kN, kN, kD, lane, wave, nwaves);
    __syncthreads();
    for (int i = tid; i < kNN; i += nth)
      ws_ptt[(size_t)t * kNN + i] = bufP[i];
    __syncthreads();
  }
}

// ---------------------------------------------------------------------------
// Kernel 2: batched Newton-Schulz pinv of p_ttm1[1..T-1] (parallel over T).
// S loaded via TDM. LDS: S | X | T1 | X2 = 256 KB per workgroup.
// ---------------------------------------------------------------------------
__global__ void pinv_batch_kernel(const float* __restrict__ ws_pttm1,
                                  float* __restrict__ ws_pinv)
{
  extern __shared__ float smem[];
  float* S  = smem;
  float* X  = smem + kNN;
  float* T1 = smem + 2 * kNN;
  float* X2 = smem + 3 * kNN;
  float* red = smem + 4 * kNN;
  const int t = blockIdx.x + 1;
  const int tid = threadIdx.x, nth = blockDim.x;
  const int lane = tid & 31, wave = tid >> 5, nwaves = nth >> 5;
  if (wave == 0) {
    tdm_load_2d(ws_pttm1 + (size_t)t * kNN, S, kN, kN, kN);
    tdm_wait();
  }
  __syncthreads();
  spd_inverse(S, X, T1, X2, red, kN, 30, tid, nth, lane, wave, nwaves);
  for (int i = tid; i < kNN; i += nth) ws_pinv[(size_t)t * kNN + i] = X[i];
}

// ---------------------------------------------------------------------------
// Kernel 3: backward RTS covariance smoother (sequential, one workgroup).
// A_t staged via TDM in a single buffer; next step's DMA overlaps GEMMs 2-4.
// ---------------------------------------------------------------------------
__global__ void cov_smooth_kernel(
    const float* __restrict__ a_g,
    const float* __restrict__ ws_ptt,
    const float* __restrict__ ws_pttm1,
    const float* __restrict__ ws_pinv,
    float* __restrict__ ws_h,     // (T,N,N), entries 0..T-2 used
    float* __restrict__ out_p)    // (T,N,N)
{
  extern __shared__ float smem[];
  float* Pn = smem;             // smoothed p_next
  float* H  = smem + kNN;
  float* Ta = smem + 2 * kNN;
  float* Tb = smem + 3 * kNN;
  float* AS = smem + 4 * kNN;   // TDM-staged A_t
  const int tid = threadIdx.x, nth = blockDim.x;
  const int lane = tid & 31, wave = tid >> 5, nwaves = nth >> 5;

  for (int i = tid; i < kNN; i += nth) {
    const float v = ws_ptt[(size_t)(kT - 1) * kNN + i];
    Pn[i] = v;
    out_p[(size_t)(kT - 1) * kNN + i] = v;
  }
  if (wave == 0) tdm_load_2d(a_g + (size_t)(kT - 2) * kNN, AS, kN, kN, kN);
  __syncthreads();
  for (int t = kT - 2; t >= 0; --t) {
    const float* pttc = ws_ptt + (size_t)t * kNN;
    if (wave == 0) tdm_wait();                    // AS = A_t ready
    __syncthreads();
    // Ta = ptt[t] @ A^T
    wmma_gemm_f32(pttc, kN, AS, kN, true, nullptr, 0, 1.f, Ta, kN,
                  kN, kN, kN, lane, wave, nwaves);
    __syncthreads();
    if (t > 0 && wave == 0)                       // prefetch A_{t-1}
      tdm_load_2d(a_g + (size_t)(t - 1) * kNN, AS, kN, kN, kN);
    // H = Ta @ pinv[t+1]
    wmma_gemm_f32(Ta, kN, ws_pinv + (size_t)(t + 1) * kNN, kN, false,
                  nullptr, 0, 1.f, H, kN, kN, kN, kN, lane, wave, nwaves);
    __syncthreads();
    for (int i = tid; i < kNN; i += nth) ws_h[(size_t)t * kNN + i] = H[i];
    // Ta = p_next - p_pred[t+1]
    for (int i = tid; i < kNN; i += nth)
      Ta[i] = Pn[i] - ws_pttm1[(size_t)(t + 1) * kNN + i];
    __syncthreads();
    // Tb = H @ Ta
    wmma_gemm_f32(H, kN, Ta, kN, false, nullptr, 0, 1.f, Tb, kN,
                  kN, kN, kN, lane, wave, nwaves);
    __syncthreads();
    // Pn = ptt[t] + Tb @ H^T
    wmma_gemm_f32(Tb, kN, H, kN, true, pttc, kN, 1.f, Pn, kN,
                  kN, kN, kN, lane, wave, nwaves);
    __syncthreads();
    for (int i = tid; i < kNN; i += nth) out_p[(size_t)t * kNN + i] = Pn[i];
    __syncthreads();
  }
}

// ---------------------------------------------------------------------------
// Kernel 4: forward mean filter. Each block owns kBT=32 batch rows, loops T.
// A_t double-buffered in LDS via TDM: DMA of A_{t+1} overlaps step t compute.
// ---------------------------------------------------------------------------
__global__ void mean_filter_kernel(
    const float* __restrict__ y_g,   // (B,T,D)
    const float* __restrict__ a_g,   // (T-1,N,N)
    const float* __restrict__ b_g,   // (D,N)
    const float* __restrict__ mu0_g, // (N)
    const float* __restrict__ ws_g,  // (T,N,D)
    float* __restrict__ ws_mu,       // (T,B,N) filtered means
    float* __restrict__ out_yhat)    // (B,T,D)
{
  extern __shared__ float smem[];
  float* mu  = smem;                 // kBT x kN
  float* muP = smem + kBT * kN;      // kBT x kN
  float* yb  = smem + 2 * kBT * kN;  // kBT x kD
  float* Ab0 = smem + 2 * kBT * kN + kBT * kD;
  float* Ab1 = Ab0 + kNN;
  const int b0 = blockIdx.x * kBT;
  const int tid = threadIdx.x, nth = blockDim.x;
  const int lane = tid & 31, wave = tid >> 5, nwaves = nth >> 5;

  if (wave == 0) tdm_load_2d(a_g, Ab1, kN, kN, kN);   // A_0 -> buf[1] (t=1)
  for (int t = 0; t < kT; ++t) {
    float* Acur = (t & 1) ? Ab1 : Ab0;
    if (t == 0) {
      for (int i = tid; i < kBT * kN; i += nth) muP[i] = mu0_g[i & (kN - 1)];
    } else {
      if (wave == 0) tdm_wait();                      // Acur ready
      __syncthreads();
      if (t + 1 < kT && wave == 0)                    // prefetch A_t
        tdm_load_2d(a_g + (size_t)t * kNN, (t & 1) ? Ab0 : Ab1, kN, kN, kN);
      // mu_pred = mu @ A^T
      wmma_gemm_f32(mu, kN, Acur, kN, true, nullptr, 0, 1.f, muP, kN,
                    kBT, kN, kN, lane, wave, nwaves);
    }
    __syncthreads();
    // yhat = mu_pred @ B^T
    wmma_gemm_f32(muP, kN, b_g, kN, true, nullptr, 0, 1.f, yb, kD,
                  kBT, kD, kN, lane, wave, nwaves);
    __syncthreads();
    for (int i = tid; i < kBT * kD; i += nth) {
      const int r = i >> 6, d = i & (kD - 1);
      const size_t gi = ((size_t)(b0 + r) * kT + t) * kD + d;
      const float yh = yb[i];
      out_yhat[gi] = yh;
      yb[i] = y_g[gi] - yh;       // innovation, in place
    }
    __syncthreads();
    // mu = mu_pred + innov @ G^T
    wmma_gemm_f32(yb, kD, ws_g + (size_t)t * kND, kD, true, muP, kN, 1.f,
                  mu, kN, kBT, kN, kD, lane, wave, nwaves);
    __syncthreads();
    for (int i = tid; i < kBT * kN; i += nth) {
      const int r = i >> 7, n = i & (kN - 1);
      ws_mu[((size_t)t * kB + b0 + r) * kN + n] = mu[i];
    }
    __syncthreads();
  }
}

// ---------------------------------------------------------------------------
// Kernel 5: backward mean smoother. A_t and H_t double-buffered via TDM
// (288 KB LDS: only possible with CDNA5's 320 KB WGP LDS pool).
// ---------------------------------------------------------------------------
__global__ void mean_smooth_kernel(
    const float* __restrict__ a_g,
    const float* __restrict__ ws_h,   // (T,N,N)
    const float* __restrict__ ws_mu,  // (T,B,N)
    float* __restrict__ out_mu)       // (B,T,N)
{
  extern __shared__ float smem[];
  float* muN = smem;                 // kBT x kN
  float* tmp = smem + kBT * kN;      // kBT x kN
  float* Ab0 = smem + 2 * kBT * kN;
  float* Ab1 = Ab0 + kNN;
  float* Hb0 = Ab1 + kNN;
  float* Hb1 = Hb0 + kNN;
  const int b0 = blockIdx.x * kBT;
  const int tid = threadIdx.x, nth = blockDim.x;
  const int lane = tid & 31, wave = tid >> 5, nwaves = nth >> 5;

  for (int i = tid; i < kBT * kN; i += nth) {
    const int r = i >> 7, n = i & (kN - 1);
    const float v = ws_mu[((size_t)(kT - 1) * kB + b0 + r) * kN + n];
    muN[i] = v;
    out_mu[((size_t)(b0 + r) * kT + kT - 1) * kN + n] = v;
  }
  if (wave == 0) {
    tdm_load_2d(a_g + (size_t)(kT - 2) * kNN, ((kT - 2) & 1) ? Ab1 : Ab0,
                kN, kN, kN);
    tdm_load_2d(ws_h + (size_t)(kT - 2) * kNN, ((kT - 2) & 1) ? Hb1 : Hb0,
                kN, kN, kN);
  }
  __syncthreads();
  for (int t = kT - 2; t >= 0; --t) {
    float* Acur = (t & 1) ? Ab1 : Ab0;
    float* Hcur = (t & 1) ? Hb1 : Hb0;
    const float* muC = ws_mu + ((size_t)t * kB + b0) * kN;
    if (wave == 0) tdm_wait();                        // Acur/Hcur ready
    __syncthreads();
    if (t > 0 && wave == 0) {                         // prefetch t-1
      tdm_load_2d(a_g + (size_t)(t - 1) * kNN, (t & 1) ? Ab0 : Ab1, kN, kN, kN);
      tdm_load_2d(ws_h + (size_t)(t - 1) * kNN, (t & 1) ? Hb0 : Hb1, kN, kN, kN);
    }
    // resid = mu_next - mu_tt @ A^T
    wmma_gemm_f32(muC, kN, Acur, kN, true, muN, kN, -1.f,
                  tmp, kN, kBT, kN, kN, lane, wave, nwaves);
    __syncthreads();
    // mu = mu_tt + resid @ H^T
    wmma_gemm_f32(tmp, kN, Hcur, kN, true, muC, kN, 1.f,
                  muN, kN, kBT, kN, kN, lane, wave, nwaves);
    __syncthreads();
    for (int i = tid; i < kBT * kN; i += nth) {
      const int r = i >> 7, n = i & (kN - 1);
      out_mu[((size_t)(b0 + r) * kT + t) * kN + n] = muN[i];
    }
    __syncthreads();
  }
}

// ---------------------------------------------------------------------------
extern "C" void kernel_launch(void* const* d_in, const int* in_sizes, int n_in,
                              void* d_out, int out_size, void* d_ws, size_t ws_size,
                              hipStream_t stream) {
  (void)in_sizes; (void)n_in; (void)out_size; (void)ws_size;
  const float* y    = (const float*)d_in[0];  // (B,T,D,1)
  const float* a    = (const float*)d_in[1];  // (T-1,N,N)
  const float* q    = (const float*)d_in[2];  // (T-1,N,N)
  const float* bmat = (const float*)d_in[3];  // (D,N)
  const float* rmat = (const float*)d_in[4];  // (D,D)
  const float* mu0  = (const float*)d_in[5];  // (N)
  const float* p0   = (const float*)d_in[6];  // (N,N)

  float* out      = (float*)d_out;
  float* out_mu   = out;                                  // (B,T,N)
  float* out_p    = out_mu   + (size_t)kB * kT * kN;      // (T,N,N)
  float* out_yhat = out_p    + (size_t)kT * kNN;          // (B,T,D)
  float* out_sig  = out_yhat + (size_t)kB * kT * kD;      // (T,D,D)

  float* ws = (float*)d_ws;   // ~218 MB total
  size_t o = 0;
  float* ws_g     = ws + o; o += (size_t)kT * kND;
  float* ws_pttm1 = ws + o; o += (size_t)kT * kNN;
  float* ws_ptt   = ws + o; o += (size_t)kT * kNN;
  float* ws_pinv  = ws + o; o += (size_t)kT * kNN;
  float* ws_h     = ws + o; o += (size_t)kT * kNN;
  float* ws_mu    = ws + o; o += (size_t)kT * kB * kN;

  cov_forward_kernel<<<1, 1024, (4 * kNN + 32) * sizeof(float), stream>>>(
      a, q, bmat, rmat, p0, ws_pttm1, ws_ptt, ws_g, out_sig);
  pinv_batch_kernel<<<kT - 1, 256, (4 * kNN + 32) * sizeof(float), stream>>>(
      ws_pttm1, ws_pinv);
  cov_smooth_kernel<<<1, 1024, (size_t)5 * kNN * sizeof(float), stream>>>(
      a, ws_ptt, ws_pttm1, ws_pinv, ws_h, out_p);
  mean_filter_kernel<<<kB / kBT, 256,
                       (2 * kBT * kN + kBT * kD + 2 * kNN) * sizeof(float),
                       stream>>>(y, a, bmat, mu0, ws_g, ws_mu, out_yhat);
  mean_smooth_kernel<<<kB / kBT, 256,
                       (2 * kBT * kN + 4 * kNN) * sizeof(float), stream>>>(
      a, ws_h, ws_mu, out_mu);
}